// Conv2dFunctionWrapped_37546604102084
// MI455X (gfx1250) — compile-verified
//
#include <hip/hip_runtime.h>

// Implicit-GEMM 3x3 valid conv on CDNA5 fp32 WMMA (V_WMMA_F32_16X16X4_F32).
// Per wave-row: 16 channels x 16 output columns, K = 9 taps padded to 12
// -> three 16x16x4 f32 WMMAs. All per-lane predication replaced by clamping
// (loads) and zero A-weights (K padding) so EXEC stays full around the WMMAs.

typedef __attribute__((ext_vector_type(2))) float v2f;
typedef __attribute__((ext_vector_type(8))) float v8f;

#define KN 16
#define H  512
#define W  512
#define OH 510
#define OW 510
#define TILE_ROWS 8
#define XT 32                                   // ceil(510/16) column tiles
#define YT ((OH + TILE_ROWS - 1) / TILE_ROWS)   // 64 row groups

__global__ __launch_bounds__(256)
void conv3x3_wmma_f32(const float* __restrict__ x,
                      const float* __restrict__ w,
                      float* __restrict__ out)
{
    const int lane = threadIdx.x & 31;
    const int wv   = threadIdx.x >> 5;
    const int tile = blockIdx.x * 8 + wv;       // 0 .. 65535

    const int xtile = tile & (XT - 1);
    const int t2    = tile >> 5;
    const int ytile = t2 & (YT - 1);
    const int b     = t2 >> 6;                  // < 32 by grid sizing

    const int x0 = xtile * 16;
    const int y0 = ytile * TILE_ROWS;
    const int n  = lane & 15;                   // N (output col) / M (channel) index
    const bool hi = (lane >= 16);

    // ---- A matrix (weights), loaded once. Step i, VGPR j: k = 4*i + j + (hi?2:0).
    // k >= 9 -> 0.0f : this zero-weight padding is what makes the B padding lanes
    // free to hold arbitrary finite data.
    v2f A[3];
#pragma unroll
    for (int i = 0; i < 3; ++i) {
#pragma unroll
        for (int j = 0; j < 2; ++j) {
            const int k = 4 * i + j + (hi ? 2 : 0);
            A[i][j] = (k < 9) ? w[n * 9 + k] : 0.0f;
        }
    }

    // ---- Clamped input columns per tap kx (only last x-tile ever clamps;
    // clamped lanes feed only output columns >= OW which are never stored).
    const int c0 = min(x0 + n + 0, W - 1);
    const int c1 = min(x0 + n + 1, W - 1);
    const int c2 = min(x0 + n + 2, W - 1);

    // Per-lane element offsets (floats) from the row-y base pointer.
    // B layout per step: lanes 0-15 hold K rows {4i, 4i+1}, lanes 16-31 {4i+2, 4i+3}.
    const int offA = hi ? c2            : c0;         // k=2 | k=0   (ky=0)
    const int offB = hi ? (W + c0)      : c1;         // k=3 | k=1
    const int offC = hi ? (2 * W + c0)  : (W + c1);   // k=6 | k=4
    const int offD = hi ? (2 * W + c1)  : (W + c2);   // k=7 | k=5
    const int offE = 2 * W + c2;                      // k=8 (all lanes; hi weights=0)

    const float* xrow = x + ((size_t)b * H + y0) * W;
    const float* pA = xrow + offA;
    const float* pB = xrow + offB;
    const float* pC = xrow + offC;
    const float* pD = xrow + offD;
    const float* pE = xrow + offE;

    const int col = x0 + n;
    const bool storeOK = (col < OW);
    // Per-lane store base: channel block (hi -> +8 planes), row y0, column col.
    float* op = out + (size_t)b * KN * OH * OW
                    + (size_t)(hi ? 8 : 0) * OH * OW
                    + (size_t)y0 * OW + col;

    const int rows = min(TILE_ROWS, OH - y0);   // wave-uniform trip count

    for (int r = 0; r < rows; ++r) {
        // ---- 5 plain loads, no predication (clause-friendly)
        const float b0 = pA[0];
        const float b1 = pB[0];
        const float b2 = pC[0];
        const float b3 = pD[0];
        const float b4 = pE[0];

        v8f c = {};
        v2f Bv0 = {b0, b1};
        v2f Bv1 = {b2, b3};
        v2f Bv2 = {b4, b4};   // second slot is K=9/11 -> A weight 0, value irrelevant
        c = __builtin_amdgcn_wmma_f32_16x16x4_f32(false, A[0], false, Bv0, (short)0, c, false, false);
        c = __builtin_amdgcn_wmma_f32_16x16x4_f32(false, A[1], false, Bv1, (short)0, c, false, false);
        c = __builtin_amdgcn_wmma_f32_16x16x4_f32(false, A[2], false, Bv2, (short)0, c, false, false);

        // ---- 8 stores: D vgpr rr -> channel m = rr + (hi?8:0); imm offsets
        // rr*OH*OW*4 <= 7,282,800 B fit the signed 24-bit IOFFSET field.
        if (storeOK) {
#pragma unroll
            for (int rr = 0; rr < 8; ++rr)
                op[(size_t)rr * OH * OW] = c[rr];
        }

        pA += W; pB += W; pC += W; pD += W; pE += W;
        op += OW;
    }
}

extern "C" void kernel_launch(void* const* d_in, const int* in_sizes, int n_in,
                              void* d_out, int out_size, void* d_ws, size_t ws_size,
                              hipStream_t stream)
{
    const float* x = (const float*)d_in[0];   // (32, 512, 512)
    const float* w = (const float*)d_in[1];   // (16, 3, 3)
    float* out = (float*)d_out;               // (32, 16, 510, 510)

    // 32 batches * 64 row-groups * 32 col-tiles = 65536 wave-tiles, 8 waves/block
    dim3 grid(65536 / 8);
    dim3 block(256);
    hipLaunchKernelGGL(conv3x3_wmma_f32, grid, block, 0, stream, x, w, out);
}